// SS_MSA_64510408786402
// MI455X (gfx1250) — compile-verified
//
#include <hip/hip_runtime.h>
#include <hip/hip_bf16.h>

typedef float v2f __attribute__((ext_vector_type(2)));
typedef float v8f __attribute__((ext_vector_type(8)));

#define DEV static __device__ __forceinline__

// D = A(16x4) * B(4x16) + C, fp32 WMMA (CDNA5 V_WMMA_F32_16X16X4_F32)
DEV v8f wmma4(v2f a, v2f b, v8f c) {
  return __builtin_amdgcn_wmma_f32_16x16x4_f32(
      false, a, false, b, (short)0, c, false, false);
}

DEV float wred_max(float v) {
  v = fmaxf(v, __shfl_xor(v, 16, 32));
  v = fmaxf(v, __shfl_xor(v, 8, 32));
  v = fmaxf(v, __shfl_xor(v, 4, 32));
  v = fmaxf(v, __shfl_xor(v, 2, 32));
  v = fmaxf(v, __shfl_xor(v, 1, 32));
  return v;
}
DEV float wred_sum(float v) {
  v += __shfl_xor(v, 16, 32);
  v += __shfl_xor(v, 8, 32);
  v += __shfl_xor(v, 4, 32);
  v += __shfl_xor(v, 2, 32);
  v += __shfl_xor(v, 1, 32);
  return v;
}

#define SCALE 0.18898223650461363f  // 28^-0.5

// ---- workspace layout (float offsets) ----
// qw/kw/vw/cat: [b][n][m][56], n=window(256), m=token-in-window(64)
static const int SZW    = 7340032;             // 8*256*64*56
static const int QW_O   = 0;
static const int KW_O   = SZW;
static const int VW_O   = 2 * SZW;
static const int CAT_O  = 3 * SZW;
static const int C1_O   = 4 * SZW;             // [br][b][16ch][16384pix]
static const int C1_SZ  = 4194304;
static const int BS_O   = C1_O + C1_SZ;        // [br][b][16ch][64 chunks]
static const int BS_SZ  = 16384;
static const int CT_O   = BS_O + BS_SZ;        // 4 arrays of 8192 (i0,i1,w0,w1)
static const int SIM2_O = CT_O + 32768;        // [b][m][256][256]

// ---------------- K1: QKV projection into windowed layout ----------------
__global__ __launch_bounds__(256) void k_qkv(
    const float* __restrict__ x, const float* __restrict__ Wq,
    const float* __restrict__ Wkv, float* __restrict__ qw,
    float* __restrict__ kw, float* __restrict__ vw) {
  int gid = blockIdx.x * 256 + threadIdx.x;   // token*168 + oc
  if (gid >= 131072 * 168) return;
  int oc = gid % 168;
  int tok = gid / 168;
  int b = tok >> 14, hw = tok & 16383, h = hw >> 7, w = hw & 127;
  const float* xr = x + (size_t)tok * 56;
  float acc = 0.f;
  if (oc < 56) {
#pragma unroll 8
    for (int c = 0; c < 56; ++c) acc += xr[c] * Wq[c * 56 + oc];
  } else {
    int o2 = oc - 56;
#pragma unroll 8
    for (int c = 0; c < 56; ++c) acc += xr[c] * Wkv[c * 112 + o2];
  }
  int n = (h >> 3) * 16 + (w >> 3);
  int m = (h & 7) * 8 + (w & 7);
  int wbase = (((b * 256 + n) * 64 + m) * 56);
  if (oc < 56) qw[wbase + oc] = acc;
  else {
    int o2 = oc - 56;
    if (o2 < 56) kw[wbase + o2] = acc;
    else         vw[wbase + o2 - 56] = acc;
  }
}

// ---------------- K2: conv1 (28ch -> 16ch, 3x3 SAME, relu) ----------------
__global__ __launch_bounds__(256) void k_conv1(
    const float* __restrict__ qw, const float* __restrict__ kw,
    const float* __restrict__ vw, const float* __restrict__ c1w,
    const float* __restrict__ c1b, float* __restrict__ c1out) {
  int gid = blockIdx.x * 256 + threadIdx.x;   // [br][b][oc][pix]
  if (gid >= 4194304) return;
  int pix = gid & 16383;
  int rest = gid >> 14;
  int oc = rest & 15; rest >>= 4;
  int b = rest & 7;
  int br = rest >> 3;
  int y = pix >> 6, xx0 = pix & 63;
  float acc = c1b[oc];
  for (int dy = 0; dy < 3; ++dy) {
    int yy = y + dy - 1;
    if ((unsigned)yy >= 256u) continue;
    for (int dx = 0; dx < 3; ++dx) {
      int xx = xx0 + dx - 1;
      if ((unsigned)xx >= 64u) continue;
      int base = ((b * 256 + yy) * 64 + xx) * 56 + br * 28;
      const float* wp = c1w + oc * 28 * 9 + dy * 3 + dx;
#pragma unroll 4
      for (int ic = 0; ic < 28; ++ic) {
        float s = qw[base + ic] + kw[base + ic] + vw[base + ic];
        acc += s * wp[ic * 9];
      }
    }
  }
  c1out[gid] = fmaxf(acc, 0.f);
}

// ------- K3: conv2 (16->16, relu) + deterministic block partial sums -------
__global__ __launch_bounds__(256) void k_conv2red(
    const float* __restrict__ c1, const float* __restrict__ c2w,
    const float* __restrict__ c2b, float* __restrict__ bsum) {
  int blk = blockIdx.x;                 // [br][b][oc][chunk64]
  int chunk = blk & 63; int rest = blk >> 6;
  int oc = rest & 15; rest >>= 4;
  int b = rest & 7;
  int br = rest >> 3;
  int pix = chunk * 256 + threadIdx.x;
  int y = pix >> 6, xx0 = pix & 63;
  float acc = c2b[oc];
  int ibase = ((br * 8 + b) * 16) * 16384;
  for (int ic = 0; ic < 16; ++ic) {
    const float* ip = c1 + ibase + ic * 16384;
    const float* wp = c2w + (oc * 16 + ic) * 9;
    for (int dy = 0; dy < 3; ++dy) {
      int yy = y + dy - 1;
      if ((unsigned)yy >= 256u) continue;
      for (int dx = 0; dx < 3; ++dx) {
        int xx = xx0 + dx - 1;
        if ((unsigned)xx >= 64u) continue;
        acc += ip[yy * 64 + xx] * wp[dy * 3 + dx];
      }
    }
  }
  float val = fmaxf(acc, 0.f);
  __shared__ float red[256];
  red[threadIdx.x] = val;
  __syncthreads();
  for (int s = 128; s; s >>= 1) {
    if (threadIdx.x < (unsigned)s) red[threadIdx.x] += red[threadIdx.x + s];
    __syncthreads();
  }
  if (threadIdx.x == 0)
    bsum[((br * 8 + b) * 16 + oc) * 64 + chunk] = red[0];
}

// -------- K4: finish means, FC shifts, precompute bilinear coord tables ----
__global__ __launch_bounds__(256) void k_shift(
    const float* __restrict__ bsum, const float* __restrict__ f1w,
    const float* __restrict__ f1b, const float* __restrict__ f2w,
    const float* __restrict__ f2b, int* __restrict__ ci0,
    int* __restrict__ ci1, float* __restrict__ cw0, float* __restrict__ cw1) {
  __shared__ float sh[2][8][2];
  int tid = threadIdx.x;
  if (tid < 16) {
    int br = tid >> 3, b = tid & 7;
    float p0 = f1b[0], p1 = f1b[1], r0 = f2b[0];
    for (int oc = 0; oc < 16; ++oc) {
      float s = 0.f;
      int base = ((br * 8 + b) * 16 + oc) * 64;
      for (int k = 0; k < 64; ++k) s += bsum[base + k];
      float mean = s * (1.f / 16384.f);
      p0 += mean * f1w[oc * 2 + 0];
      p1 += mean * f1w[oc * 2 + 1];
      r0 += mean * f2w[oc];
    }
    sh[br][b][0] = r0 * tanhf(p0);
    sh[br][b][1] = r0 * tanhf(p1);
  }
  __syncthreads();
  for (int e = tid; e < 8192; e += 256) {   // [br][b][axis][t(256)]
    int t = e & 255;
    int axis = (e >> 8) & 1;
    int b = (e >> 9) & 7;
    int br = e >> 12;
    int size = (br == 0) ? 64 : 256;
    int i0 = 0, i1 = 0; float w0 = 0.f, w1 = 0.f;
    if (t < size) {
      float s = sh[br][b][axis];
      float span = (float)(size - 1);
      float g = -1.f + 2.f * (float)t / span;
      float c = (g + s * 2.f / (float)size + 1.f) * 0.5f * span;
      c = fmodf(fabsf(c), 2.f * span);
      if (c > span) c = 2.f * span - c;
      float fl = fminf(fmaxf(floorf(c), 0.f), span);
      i0 = (int)fl;
      i1 = (i0 + 1 < size - 1) ? (i0 + 1) : (size - 1);
      w1 = c - fl;
      w0 = 1.f - w1;
    }
    ci0[e] = i0; ci1[e] = i1; cw0[e] = w0; cw1[e] = w1;
  }
}

// ---- K5: branch-1 attention (per (b, window)): WMMA QK^T, shift, softmax, WMMA PV
__global__ __launch_bounds__(256) void k_attn1(
    const float* __restrict__ qw, const float* __restrict__ kw,
    const float* __restrict__ vw, const float* __restrict__ pos1,
    const int* __restrict__ ci0, const int* __restrict__ ci1,
    const float* __restrict__ cw0, const float* __restrict__ cw1,
    float* __restrict__ cat) {
  __shared__ float sim[64][64];
  __shared__ float att[64][64];
  int b = blockIdx.x >> 8, n = blockIdx.x & 255;
  int tid = threadIdx.x, wv = tid >> 5, ln = tid & 31, lm = ln & 15, lh = ln >> 4;
  int base = ((b * 256 + n) * 64) * 56;

  // sim = SCALE*q@k^T + pos1   (16 tiles, 2 per wave)
  for (int tt = 0; tt < 2; ++tt) {
    int ti = wv * 2 + tt;
    int i0 = (ti >> 2) << 4, j0 = (ti & 3) << 4;
    v8f c;
#pragma unroll
    for (int r = 0; r < 8; ++r) c[r] = pos1[(i0 + r + 8 * lh) * 64 + j0 + lm];
#pragma unroll
    for (int kb = 0; kb < 28; kb += 4) {
      int d = kb + 2 * lh;
      v2f a, bb;
      a.x = SCALE * qw[base + (i0 + lm) * 56 + d];
      a.y = SCALE * qw[base + (i0 + lm) * 56 + d + 1];
      bb.x = kw[base + (j0 + lm) * 56 + d];
      bb.y = kw[base + (j0 + lm) * 56 + d + 1];
      c = wmma4(a, bb, c);
    }
#pragma unroll
    for (int r = 0; r < 8; ++r) sim[i0 + r + 8 * lh][j0 + lm] = c[r];
  }
  __syncthreads();

  // grid_shift + softmax (8 rows per wave)
  int cbx = ((0 * 8 + b) * 2 + 0) * 256;  // x/j tables (size 64)
  int cby = ((0 * 8 + b) * 2 + 1) * 256;  // y/i tables
  for (int rr = 0; rr < 8; ++rr) {
    int i = wv * 8 + rr;
    int y0 = ci0[cby + i], y1 = ci1[cby + i];
    float wy0 = cw0[cby + i], wy1 = cw1[cby + i];
    int ip = (i + 63) & 63;
    float v[2];
#pragma unroll
    for (int t = 0; t < 2; ++t) {
      int j = ln + 32 * t;
      int x0 = ci0[cbx + j], x1 = ci1[cbx + j];
      float wx0 = cw0[cbx + j], wx1 = cw1[cbx + j];
      int jp = (j + 63) & 63;
      v[t] = sim[ip][jp] +
             wy0 * (wx0 * sim[y0][x0] + wx1 * sim[y0][x1]) +
             wy1 * (wx0 * sim[y1][x0] + wx1 * sim[y1][x1]);
    }
    float mx = wred_max(fmaxf(v[0], v[1]));
    float e0 = __expf(v[0] - mx), e1 = __expf(v[1] - mx);
    float inv = 1.f / wred_sum(e0 + e1);
    att[i][ln] = e0 * inv;
    att[i][ln + 32] = e1 * inv;
  }
  __syncthreads();

  // out = att(64x64) @ v(64x28) ; 8 tiles (pad N to 32), 1 per wave
  int m0 = (wv & 3) << 4, j0 = (wv >> 2) << 4;
  int col = j0 + lm;
  bool ok = col < 28;
  v8f c;
#pragma unroll
  for (int r = 0; r < 8; ++r) c[r] = 0.f;
  for (int kb = 0; kb < 64; kb += 4) {
    int kr = kb + 2 * lh;
    v2f a, bb;
    a.x = att[m0 + lm][kr];
    a.y = att[m0 + lm][kr + 1];
    bb.x = ok ? vw[base + kr * 56 + col] : 0.f;
    bb.y = ok ? vw[base + (kr + 1) * 56 + col] : 0.f;
    c = wmma4(a, bb, c);
  }
#pragma unroll
  for (int r = 0; r < 8; ++r)
    if (col < 28) cat[base + (m0 + r + 8 * lh) * 56 + col] = c[r];
}

// ---- K6: branch-2 raw logits (per (b, head m)): 256x256 via WMMA ----
__global__ __launch_bounds__(256) void k_sim2(
    const float* __restrict__ qw, const float* __restrict__ kw,
    const float* __restrict__ pos2, float* __restrict__ sim2) {
  int b = blockIdx.x >> 6, m = blockIdx.x & 63;
  int tid = threadIdx.x, wv = tid >> 5, ln = tid & 31, lm = ln & 15, lh = ln >> 4;
  int sbase = (b * 64 + m) * 65536;
  for (int tt = 0; tt < 32; ++tt) {
    int ti = wv + tt * 8;
    int i0 = (ti >> 4) << 4, j0 = (ti & 15) << 4;
    v8f c;
#pragma unroll
    for (int r = 0; r < 8; ++r) c[r] = pos2[(i0 + r + 8 * lh) * 256 + j0 + lm];
    int arow = ((b * 256 + i0 + lm) * 64 + m) * 56 + 28;
    int brow = ((b * 256 + j0 + lm) * 64 + m) * 56 + 28;
#pragma unroll
    for (int kb = 0; kb < 28; kb += 4) {
      int d = kb + 2 * lh;
      v2f a, bb;
      a.x = SCALE * qw[arow + d];
      a.y = SCALE * qw[arow + d + 1];
      bb.x = kw[brow + d];
      bb.y = kw[brow + d + 1];
      c = wmma4(a, bb, c);
    }
#pragma unroll
    for (int r = 0; r < 8; ++r)
      sim2[sbase + (i0 + r + 8 * lh) * 256 + j0 + lm] = c[r];
  }
}

// ---- K7: branch-2 shift + softmax + PV (per (b, m, 16-row tile)) ----
__global__ __launch_bounds__(256) void k_attn2(
    const float* __restrict__ vw, const float* __restrict__ sim2,
    const int* __restrict__ ci0, const int* __restrict__ ci1,
    const float* __restrict__ cw0, const float* __restrict__ cw1,
    float* __restrict__ cat) {
  __shared__ float att[16][256];
  int bx = blockIdx.x;
  int b = bx >> 10, m = (bx >> 4) & 63, ib = bx & 15;
  int tid = threadIdx.x, wv = tid >> 5, ln = tid & 31, lm = ln & 15, lh = ln >> 4;
  const float* raw = sim2 + (b * 64 + m) * 65536;
  int cbx = ((1 * 8 + b) * 2 + 0) * 256;
  int cby = ((1 * 8 + b) * 2 + 1) * 256;
  for (int rr = wv * 2; rr < wv * 2 + 2; ++rr) {
    int i = ib * 16 + rr;
    int y0 = ci0[cby + i], y1 = ci1[cby + i];
    float wy0 = cw0[cby + i], wy1 = cw1[cby + i];
    int ip = (i + 255) & 255;
    float v[8];
    float mx = -3.4e38f;
#pragma unroll
    for (int t = 0; t < 8; ++t) {
      int j = ln + 32 * t;
      int x0 = ci0[cbx + j], x1 = ci1[cbx + j];
      float wx0 = cw0[cbx + j], wx1 = cw1[cbx + j];
      int jp = (j + 255) & 255;
      v[t] = raw[ip * 256 + jp] +
             wy0 * (wx0 * raw[y0 * 256 + x0] + wx1 * raw[y0 * 256 + x1]) +
             wy1 * (wx0 * raw[y1 * 256 + x0] + wx1 * raw[y1 * 256 + x1]);
      mx = fmaxf(mx, v[t]);
    }
    mx = wred_max(mx);
    float s = 0.f;
#pragma unroll
    for (int t = 0; t < 8; ++t) { v[t] = __expf(v[t] - mx); s += v[t]; }
    float inv = 1.f / wred_sum(s);
#pragma unroll
    for (int t = 0; t < 8; ++t) att[rr][ln + 32 * t] = v[t] * inv;
  }
  __syncthreads();
  if (wv < 2) {  // whole waves only: EXEC all-ones inside
    int j0 = wv << 4;
    int col = j0 + lm;
    bool ok = col < 28;
    v8f c;
#pragma unroll
    for (int r = 0; r < 8; ++r) c[r] = 0.f;
    for (int kb = 0; kb < 256; kb += 4) {
      int kr = kb + 2 * lh;
      v2f a, bb;
      a.x = att[lm][kr];
      a.y = att[lm][kr + 1];
      bb.x = ok ? vw[((b * 256 + kr) * 64 + m) * 56 + 28 + col] : 0.f;
      bb.y = ok ? vw[((b * 256 + kr + 1) * 64 + m) * 56 + 28 + col] : 0.f;
      c = wmma4(a, bb, c);
    }
#pragma unroll
    for (int r = 0; r < 8; ++r)
      if (col < 28) {
        int i = ib * 16 + r + 8 * lh;
        cat[((b * 256 + i) * 64 + m) * 56 + 28 + col] = c[r];
      }
  }
}

// ---- K8: output projection + un-window to NHWC ----
__global__ __launch_bounds__(256) void k_proj(
    const float* __restrict__ cat, const float* __restrict__ Wout,
    const float* __restrict__ bout, float* __restrict__ out) {
  int gid = blockIdx.x * 256 + threadIdx.x;
  if (gid >= 131072 * 56) return;
  int oc = gid % 56;
  int tok = gid / 56;
  int b = tok >> 14, n = (tok >> 6) & 255, m = tok & 63;
  float acc = bout[oc];
  const float* cr = cat + tok * 56;
#pragma unroll 8
  for (int c = 0; c < 56; ++c) acc += cr[c] * Wout[c * 56 + oc];
  int h = (n >> 4) * 8 + (m >> 3);
  int w = (n & 15) * 8 + (m & 7);
  out[((b * 128 + h) * 128 + w) * 56 + oc] = acc;
}

extern "C" void kernel_launch(void* const* d_in, const int* in_sizes, int n_in,
                              void* d_out, int out_size, void* d_ws,
                              size_t ws_size, hipStream_t stream) {
  (void)in_sizes; (void)n_in; (void)out_size; (void)ws_size;
  const float* x    = (const float*)d_in[0];
  const float* pos1 = (const float*)d_in[1];
  const float* pos2 = (const float*)d_in[2];
  const float* Wq   = (const float*)d_in[3];
  const float* Wkv  = (const float*)d_in[4];
  const float* Wout = (const float*)d_in[5];
  const float* bout = (const float*)d_in[6];
  const float* c1w  = (const float*)d_in[7];
  const float* c1b  = (const float*)d_in[8];
  const float* c2w  = (const float*)d_in[9];
  const float* c2b  = (const float*)d_in[10];
  const float* f1w  = (const float*)d_in[11];
  const float* f1b  = (const float*)d_in[12];
  const float* f2w  = (const float*)d_in[13];
  const float* f2b  = (const float*)d_in[14];

  float* ws  = (float*)d_ws;
  int*   wsi = (int*)d_ws;
  float* qw    = ws + QW_O;
  float* kw    = ws + KW_O;
  float* vw    = ws + VW_O;
  float* cat   = ws + CAT_O;
  float* c1out = ws + C1_O;
  float* bsum  = ws + BS_O;
  int*   ci0   = wsi + CT_O;
  int*   ci1   = wsi + CT_O + 8192;
  float* cw0   = ws + CT_O + 16384;
  float* cw1   = ws + CT_O + 24576;
  float* sim2  = ws + SIM2_O;
  float* out   = (float*)d_out;

  k_qkv<<<86016, 256, 0, stream>>>(x, Wq, Wkv, qw, kw, vw);
  k_conv1<<<16384, 256, 0, stream>>>(qw, kw, vw, c1w, c1b, c1out);
  k_conv2red<<<16384, 256, 0, stream>>>(c1out, c2w, c2b, bsum);
  k_shift<<<1, 256, 0, stream>>>(bsum, f1w, f1b, f2w, f2b, ci0, ci1, cw0, cw1);
  k_attn1<<<2048, 256, 0, stream>>>(qw, kw, vw, pos1, ci0, ci1, cw0, cw1, cat);
  k_sim2<<<512, 256, 0, stream>>>(qw, kw, pos2, sim2);
  k_attn2<<<8192, 256, 0, stream>>>(vw, sim2, ci0, ci1, cw0, cw1, cat);
  k_proj<<<28672, 256, 0, stream>>>(cat, Wout, bout, out);
}